// CRF_11647951307120
// MI455X (gfx1250) — compile-verified
//
#include <hip/hip_runtime.h>

typedef float v2f __attribute__((ext_vector_type(2)));
typedef float v4f __attribute__((ext_vector_type(4)));
typedef float v8f __attribute__((ext_vector_type(8)));

#define B_ 512
#define S_ 1024
#define L_ 32

#define LOG2E_F 1.44269504088896340736f
#define LN2_F   0.69314718055994530942f

__device__ __forceinline__ float wave_max32(float v) {
#pragma unroll
  for (int off = 16; off >= 1; off >>= 1)
    v = fmaxf(v, __shfl_xor(v, off, 32));
  return v;
}

__device__ __forceinline__ float wave_sum32(float v) {
#pragma unroll
  for (int off = 16; off >= 1; off >>= 1)
    v += __shfl_xor(v, off, 32);
  return v;
}

__device__ __forceinline__ float readfirstlane_f32(float v) {
  return __int_as_float(__builtin_amdgcn_readfirstlane(__float_as_int(v)));
}

// ---------------------------------------------------------------------------
// Linear-space forward scan: one wave per batch, lane j <-> state j.
//   p_t = (p_{t-1} . expT) * exp(emit_t) * 2^{-k}
// Invariant: exp(alpha_t[j]) = p_t[j] * 2^{Ktot} * exp(m0)  (exact).
// The 2^{-k} rescale exponent is harvested from readfirstlane(p) one step
// late (SALU bit ops), and exp(emit) comes from the prefetched emission, so
// neither exp, log, nor any reduction sits on the per-step critical chain:
//   chain = v_mul -> ds_store/ds_load_b128 -> WMMA(depth 2) -> add tree.
// K-dimension permutation sigma(4kc+r+2h)=8(kc>>1)+4h+2(kc&1)+r makes each
// chunk-pair's A fragment 4 contiguous LDS floats (ds_load_b128 broadcast);
// expT rows are reordered to match at init.
// ---------------------------------------------------------------------------
__global__ __launch_bounds__(128) void crf_forward_kernel(
    const float* __restrict__ emissions,
    const float* __restrict__ transitions,
    float* __restrict__ part_out) {
  __shared__ float expT[L_ * L_];
  __shared__ float evec[4][L_];

  const int lane  = threadIdx.x & 31;
  const int wave  = threadIdx.x >> 5;
  const int batch = blockIdx.x * 4 + wave;

  // expT = exp(T), built cooperatively.
  for (int i = threadIdx.x; i < L_ * L_; i += 128)
    expT[i] = __builtin_amdgcn_exp2f(transitions[i] * LOG2E_F);
  __syncthreads();

  // B tiles with sigma-permuted K rows (4x16 f32 layout: VGPR r, lane half h
  // holds row K = 4kc+r+2h; actual source state = sigma(K)).
  const int h4   = (lane < 16) ? 0 : 4;  // 4*h
  const int jcol = lane & 15;
  float bmat[8][2][2];
#pragma unroll
  for (int kc = 0; kc < 8; ++kc) {
#pragma unroll
    for (int nt = 0; nt < 2; ++nt) {
#pragma unroll
      for (int r = 0; r < 2; ++r) {
        const int state = 8 * (kc >> 1) + h4 + 2 * (kc & 1) + r;
        bmat[kc][nt][r] = expT[state * L_ + nt * 16 + jcol];
      }
    }
  }

  const float* em = emissions + (size_t)batch * S_ * L_ + lane;
  float* eslot    = evec[wave];

  // p0 = exp(emit0 - m0), wave-uniform shift m0 keeps p0 in range.
  const float emit0 = em[0];
  const float m0    = readfirstlane_f32(emit0);
  float p           = __builtin_amdgcn_exp2f((emit0 - m0) * LOG2E_F);
  eslot[lane] = p;

  float emit_next = em[L_];
  float cscale    = 1.0f;  // 2^{-kpend}, applied this step
  int   kpend     = 0;     // exponent measured last step, applied this step
  int   ktot      = 0;     // sum of applied exponents

  for (int t = 1; t < S_; ++t) {
    // ---- off-critical-chain work -------------------------------------
    const float expE = __builtin_amdgcn_exp2f(emit_next * LOG2E_F);
    const int tnext  = (t + 1 < S_) ? (t + 1) : (S_ - 1);
    emit_next = em[(size_t)tnext * L_];               // prefetch
    const float cmul = expE * cscale;
    ktot += kpend;

    // ---- critical chain: 4 x ds_load_b128 broadcast of p fragments ---
    v8f a0 = {}, a1 = {}, a2 = {}, a3 = {};   // tile 0 (N=0..15), depth-2 chains
    v8f b0 = {}, b1 = {}, b2 = {}, b3 = {};   // tile 1 (N=16..31)
#pragma unroll
    for (int P = 0; P < 4; ++P) {
      const v4f q = *(const v4f*)&eslot[8 * P + h4];
      const v2f alo = {q.x, q.y};  // chunk 2P
      const v2f ahi = {q.z, q.w};  // chunk 2P+1
      const int kc0 = 2 * P, kc1 = 2 * P + 1;
      const v2f B00 = {bmat[kc0][0][0], bmat[kc0][0][1]};
      const v2f B01 = {bmat[kc0][1][0], bmat[kc0][1][1]};
      const v2f B10 = {bmat[kc1][0][0], bmat[kc1][0][1]};
      const v2f B11 = {bmat[kc1][1][0], bmat[kc1][1][1]};
      switch (P) {
        case 0:
          a0 = __builtin_amdgcn_wmma_f32_16x16x4_f32(false, alo, false, B00, (short)0, a0, false, false);
          b0 = __builtin_amdgcn_wmma_f32_16x16x4_f32(false, alo, false, B01, (short)0, b0, false, false);
          a1 = __builtin_amdgcn_wmma_f32_16x16x4_f32(false, ahi, false, B10, (short)0, a1, false, false);
          b1 = __builtin_amdgcn_wmma_f32_16x16x4_f32(false, ahi, false, B11, (short)0, b1, false, false);
          break;
        case 1:
          a2 = __builtin_amdgcn_wmma_f32_16x16x4_f32(false, alo, false, B00, (short)0, a2, false, false);
          b2 = __builtin_amdgcn_wmma_f32_16x16x4_f32(false, alo, false, B01, (short)0, b2, false, false);
          a3 = __builtin_amdgcn_wmma_f32_16x16x4_f32(false, ahi, false, B10, (short)0, a3, false, false);
          b3 = __builtin_amdgcn_wmma_f32_16x16x4_f32(false, ahi, false, B11, (short)0, b3, false, false);
          break;
        case 2:
          a0 = __builtin_amdgcn_wmma_f32_16x16x4_f32(false, alo, false, B00, (short)0, a0, false, false);
          b0 = __builtin_amdgcn_wmma_f32_16x16x4_f32(false, alo, false, B01, (short)0, b0, false, false);
          a1 = __builtin_amdgcn_wmma_f32_16x16x4_f32(false, ahi, false, B10, (short)0, a1, false, false);
          b1 = __builtin_amdgcn_wmma_f32_16x16x4_f32(false, ahi, false, B11, (short)0, b1, false, false);
          break;
        default:
          a2 = __builtin_amdgcn_wmma_f32_16x16x4_f32(false, alo, false, B00, (short)0, a2, false, false);
          b2 = __builtin_amdgcn_wmma_f32_16x16x4_f32(false, alo, false, B01, (short)0, b2, false, false);
          a3 = __builtin_amdgcn_wmma_f32_16x16x4_f32(false, ahi, false, B10, (short)0, a3, false, false);
          b3 = __builtin_amdgcn_wmma_f32_16x16x4_f32(false, ahi, false, B11, (short)0, b3, false, false);
      }
    }
    // Only D VGPR0 is consumed (rows replicated):
    // lanes 0-15 -> tile0 column `lane`; lanes 16-31 -> tile1 column `lane`.
    const float y0 = (a0[0] + a1[0]) + (a2[0] + a3[0]);
    const float y1 = (b0[0] + b1[0]) + (b2[0] + b3[0]);
    const float y  = (lane < 16) ? y0 : y1;

    p = y * cmul;          // emission + delayed 2^{-k} rescale, one v_mul
    eslot[lane] = p;       // feeds next iteration's A fragments

    // ---- off-chain: harvest next rescale exponent (SALU bit ops) ------
    const float rf = readfirstlane_f32(p);
    kpend  = ((__float_as_int(rf) >> 23) & 255) - 127;
    cscale = __int_as_float((127 - kpend) << 23);
  }

  // log-partition: logZ = ln(sum_j p) + Ktot*ln2 + m0  (p ~ 2^0, safe sum)
  const float s = wave_sum32(p);
  if (lane == 0)
    part_out[batch] = (__builtin_amdgcn_logf(s) + (float)ktot) * LN2_F + m0;
}

// ---------------------------------------------------------------------------
// Gold-path score: unary gather + transition gather, one block per batch.
// ---------------------------------------------------------------------------
__global__ __launch_bounds__(1024) void crf_score_kernel(
    const float* __restrict__ emissions,
    const int* __restrict__ tags,
    const float* __restrict__ transitions,
    float* __restrict__ score_out) {
  __shared__ float red[32];
  const int b = blockIdx.x;
  const int t = threadIdx.x;
  const size_t idx = (size_t)b * S_ + t;

  const int tag = tags[idx];
  float v = emissions[idx * L_ + tag];
  if (t < S_ - 1) v += transitions[tag * L_ + tags[idx + 1]];

  v = wave_sum32(v);
  const int lane = t & 31, wv = t >> 5;
  if (lane == 0) red[wv] = v;
  __syncthreads();
  if (wv == 0) {
    float x = red[lane];
    x = wave_sum32(x);
    if (lane == 0) score_out[b] = x;
  }
}

// ---------------------------------------------------------------------------
// loss = sum_b part[b] - sum_b score[b]  (deterministic tree reduction)
// ---------------------------------------------------------------------------
__global__ __launch_bounds__(512) void crf_finalize_kernel(
    const float* __restrict__ part,
    const float* __restrict__ score,
    float* __restrict__ out) {
  __shared__ float red[16];
  const int i = threadIdx.x;  // 512 threads, one per batch
  float v = part[i] - score[i];
  v = wave_sum32(v);
  const int lane = i & 31, wv = i >> 5;
  if (lane == 0) red[wv] = v;
  __syncthreads();
  if (wv == 0) {
    float x = (lane < 16) ? red[lane] : 0.0f;
    x = wave_sum32(x);
    if (lane == 0) out[0] = x;
  }
}

extern "C" void kernel_launch(void* const* d_in, const int* in_sizes, int n_in,
                              void* d_out, int out_size, void* d_ws, size_t ws_size,
                              hipStream_t stream) {
  const float* emissions   = (const float*)d_in[0];
  const int*   tags        = (const int*)d_in[1];
  const float* transitions = (const float*)d_in[2];

  float* ws    = (float*)d_ws;
  float* part  = ws;         // 512 floats: per-batch log-partition
  float* score = ws + B_;    // 512 floats: per-batch gold score

  crf_score_kernel<<<B_, S_, 0, stream>>>(emissions, tags, transitions, score);
  crf_forward_kernel<<<B_ / 4, 128, 0, stream>>>(emissions, transitions, part);
  crf_finalize_kernel<<<1, B_, 0, stream>>>(part, score, (float*)d_out);
}